// SAGELayer_82678120448015
// MI455X (gfx1250) — compile-verified
//
#include <hip/hip_runtime.h>

#define B_   64
#define L_   64
#define D_   512
#define FNB  32
#define H_   512

typedef __attribute__((ext_vector_type(2))) float v2f;
typedef __attribute__((ext_vector_type(8))) float v8f;
typedef unsigned int u32x4 __attribute__((ext_vector_type(4)));
typedef int i32x8 __attribute__((ext_vector_type(8)));
typedef int i32x4 __attribute__((ext_vector_type(4)));

#if defined(__has_builtin)
#if __has_builtin(__builtin_amdgcn_tensor_load_to_lds) && \
    __has_builtin(__builtin_amdgcn_s_wait_tensorcnt)
#define USE_TDM 1
#endif
#endif

// ---------------------------------------------------------------------------
// Stage 1: build A_self[2112,512] and A_aggr[2112,512].
//   rows 0..2047  : r=(b,f)  A_self = x[b, idx1[b,f]]   A_aggr = mean_j x[b, idx2[b,f,j]]
//   rows 2048..2111: r=b     A_self = x[b, 0]           A_aggr = mean_j x[b, idx1[b,j]]
// ---------------------------------------------------------------------------
__global__ __launch_bounds__(256) void gather_mean_kernel(
    const float* __restrict__ x, const int* __restrict__ idx1,
    const int* __restrict__ idx2, float* __restrict__ Aself,
    float* __restrict__ Aaggr)
{
  const int row = blockIdx.x;
  const int t   = threadIdx.x;

  const float* self_src;
  const int*   nbr;
  int b;
  if (row < B_ * FNB) {
    b = row >> 5;
    const int f = row & 31;
    self_src = x + (size_t)(b * L_ + idx1[b * FNB + f]) * D_;
    nbr      = idx2 + (size_t)b * FNB * FNB + f * FNB;
  } else {
    b = row - B_ * FNB;
    self_src = x + (size_t)(b * L_) * D_;
    nbr      = idx1 + (size_t)b * FNB;
  }

  // Block-uniform neighbor indices -> SGPRs.
  int nb[32];
#pragma unroll
  for (int j = 0; j < 32; ++j) nb[j] = nbr[j];

  const float* xb = x + (size_t)b * L_ * D_;
  for (int c = t; c < D_; c += 256) {
    Aself[(size_t)row * D_ + c] = self_src[c];
    float s = 0.f;
#pragma unroll
    for (int j = 0; j < 32; ++j) s += xb[(size_t)nb[j] * D_ + c];
    Aaggr[(size_t)row * D_ + c] = s * (1.0f / 32.0f);
  }
}

// ---------------------------------------------------------------------------
// Stage 2/4: out[M,512] = act(A1 @ W1 + A2 @ W2 + bias), K = 512 per matrix.
// Block: 256 threads = 8 waves, 64x64 output tile. Wave (wm,wn) -> 16x32 slice.
// f32 WMMA 16x16x4. A tile staged by TDM (tensor_load_to_lds) with HW LDS
// padding reproducing the 36-dword row stride; B tile staged transposed so
// each B fragment is a single aligned ds_load_b64.
// ---------------------------------------------------------------------------
__global__ __launch_bounds__(256) void sage_gemm_kernel(
    const float* __restrict__ A1, const float* __restrict__ A2,
    const float* __restrict__ W1, const float* __restrict__ W2,
    const float* __restrict__ bias, float* __restrict__ out,
    int leaky)
{
  __shared__ float As[64][36];    // 64 rows x 32 K, stride 36 (TDM pad 32+4 dwords)
  __shared__ float BsT[64][34];   // transposed: [n][k], stride 34

  const int t    = threadIdx.x;
  const int lane = t & 31;
  const int wave = t >> 5;       // 0..7
  const int wm   = wave >> 1;    // 0..3  -> m sub-tile
  const int wn   = wave & 1;     // 0..1  -> n sub-tile (x2 16-wide)
  const int half = lane >> 4;    // 0..1
  const int l15  = lane & 15;

  const int n0 = blockIdx.x * 64;
  const int m0 = blockIdx.y * 64;

#ifdef USE_TDM
  const unsigned lds_as_base = (unsigned)(size_t)&As[0][0];
#endif

  v8f acc0 = {};
  v8f acc1 = {};

  for (int src = 0; src < 2; ++src) {
    const float* A = src ? A2 : A1;
    const float* W = src ? W2 : W1;
    for (int kb = 0; kb < D_ / 32; ++kb) {
      const int k0 = kb * 32;
      __syncthreads();   // previous chunk fully consumed before overwrite

#ifdef USE_TDM
      if (wave == 0) {
        // --- TDM descriptor: 2D tile 32(K) x 64(M) f32, global row stride 512,
        // --- LDS pad: after every 32 dwords insert 4 dwords -> stride 36.
        unsigned long long ga =
            (unsigned long long)(const void*)(A + (size_t)m0 * D_ + k0);
        u32x4 g0;
        g0.x = 1u;                                        // count=1, load descr
        g0.y = lds_as_base;                               // lds_addr (bytes)
        g0.z = (unsigned)(ga & 0xFFFFFFFFu);              // global_addr[31:0]
        g0.w = (unsigned)((ga >> 32) & 0x01FFFFFFu)       // global_addr[56:32]
             | (2u << 30);                                // type=2 ("image")
        i32x8 g1;
        g1[0] = (2 << 16)      // data_size = 4B
              | (1 << 20)      // pad_enable
              | (4 << 22)      // pad_interval code 4 -> every 32 dwords
              | (3 << 25);     // pad_amount  code 3 -> 4 dwords
        g1[1] = (int)(512u << 16);   // tensor_dim0 = 512 (bits 79:48, low half)
        g1[2] = (int)(64u << 16);    // tensor_dim0 hi = 0 ; tensor_dim1 = 64
        g1[3] = (int)(32u << 16);    // tensor_dim1 hi = 0 ; tile_dim0 = 32
        g1[4] = 64;                  // tile_dim1 = 64, tile_dim2 = 0 (2D)
        g1[5] = 512;                 // tensor_dim0_stride[31:0] = 512 elems
        g1[6] = 0;                   // stride0 hi / stride1 lo
        g1[7] = 0;
        const i32x4 z4 = {0, 0, 0, 0};
        const i32x8 z8 = {0, 0, 0, 0, 0, 0, 0, 0};
        __builtin_amdgcn_tensor_load_to_lds(g0, g1, z4, z4, z8, 0);
      }
#else
#pragma unroll
      for (int i = 0; i < 8; ++i) {          // A tile: 64x32
        const int idx = t + i * 256;
        const int r = idx >> 5, c = idx & 31;
        As[r][c] = A[(size_t)(m0 + r) * D_ + k0 + c];
      }
#endif

#pragma unroll
      for (int i = 0; i < 8; ++i) {          // B tile: 32(K) x 64(N), transposed
        const int idx = t + i * 256;
        const int r = idx >> 6, c = idx & 63;   // r = k row, c = n col
        BsT[c][r] = W[(size_t)(k0 + r) * H_ + n0 + c];
      }

#ifdef USE_TDM
      if (wave == 0) __builtin_amdgcn_s_wait_tensorcnt(0);
#endif
      __syncthreads();

#pragma unroll
      for (int kk = 0; kk < 32; kk += 4) {
        const int ak = kk + 2 * half;   // lanes 0-15: K=kk,kk+1 ; 16-31: kk+2,kk+3
        const v2f a  = *(const v2f*)&As[wm * 16 + l15][ak];
        const v2f b0 = *(const v2f*)&BsT[wn * 32 + l15][ak];
        const v2f b1 = *(const v2f*)&BsT[wn * 32 + 16 + l15][ak];
        acc0 = __builtin_amdgcn_wmma_f32_16x16x4_f32(
            false, a, false, b0, (short)0, acc0, false, false);
        acc1 = __builtin_amdgcn_wmma_f32_16x16x4_f32(
            false, a, false, b1, (short)0, acc1, false, false);
      }
    }
  }

  const int col0 = n0 + wn * 32 + l15;
  const int col1 = col0 + 16;
  const float bv0 = bias[col0];
  const float bv1 = bias[col1];
#pragma unroll
  for (int r = 0; r < 8; ++r) {
    const int row = m0 + wm * 16 + r + half * 8;   // C/D layout: VGPR r, half split
    float v0 = acc0[r] + bv0;
    float v1 = acc1[r] + bv1;
    if (leaky) {
      v0 = v0 > 0.f ? v0 : 0.01f * v0;
      v1 = v1 > 0.f ? v1 : 0.01f * v1;
    }
    out[(size_t)row * H_ + col0] = v0;
    out[(size_t)row * H_ + col1] = v1;
  }
}

// ---------------------------------------------------------------------------
// Stage 3: mh1[b,c] = mean over f of Y[b*32+f, c]
// ---------------------------------------------------------------------------
__global__ __launch_bounds__(256) void mean_rows_kernel(
    const float* __restrict__ Y, float* __restrict__ mh1)
{
  const int b = blockIdx.x;
  const int t = threadIdx.x;
  for (int c = t; c < H_; c += 256) {
    float s = 0.f;
#pragma unroll
    for (int f = 0; f < 32; ++f) s += Y[(size_t)(b * 32 + f) * H_ + c];
    mh1[(size_t)b * H_ + c] = s * (1.0f / 32.0f);
  }
}

extern "C" void kernel_launch(void* const* d_in, const int* in_sizes, int n_in,
                              void* d_out, int out_size, void* d_ws, size_t ws_size,
                              hipStream_t stream) {
  (void)in_sizes; (void)n_in; (void)out_size; (void)ws_size;
  const float* x       = (const float*)d_in[0];
  const int*   idx1    = (const int*)d_in[1];
  const int*   idx2    = (const int*)d_in[2];
  const float* W_aggr0 = (const float*)d_in[3];
  const float* b_aggr0 = (const float*)d_in[4];
  const float* W_self0 = (const float*)d_in[5];
  const float* W_aggr1 = (const float*)d_in[6];
  const float* b_aggr1 = (const float*)d_in[7];
  const float* W_self1 = (const float*)d_in[8];
  float*       out     = (float*)d_out;

  const int Mbig = B_ * FNB + B_;   // 2112 rows: [h1 inputs | h0 inputs]
  float* ws    = (float*)d_ws;
  float* Aself = ws;                              // 2112*512
  float* Aaggr = Aself + (size_t)Mbig * D_;       // 2112*512
  float* Y     = Aaggr + (size_t)Mbig * D_;       // 2112*512 (h1 | h0)
  float* mh1   = Y     + (size_t)Mbig * H_;       // 64*512

  gather_mean_kernel<<<Mbig, 256, 0, stream>>>(x, idx1, idx2, Aself, Aaggr);

  sage_gemm_kernel<<<dim3(H_ / 64, Mbig / 64), 256, 0, stream>>>(
      Aself, Aaggr, W_self0, W_aggr0, b_aggr0, Y, /*leaky=*/1);

  mean_rows_kernel<<<B_, 256, 0, stream>>>(Y, mh1);

  sage_gemm_kernel<<<dim3(H_ / 64, 1), 256, 0, stream>>>(
      Y + (size_t)(B_ * FNB) * H_, mh1, W_self1, W_aggr1, b_aggr1, out, /*leaky=*/0);
}